// BIGA_44976897524676
// MI455X (gfx1250) — compile-verified
//
#include <hip/hip_runtime.h>
#include <hip/hip_bf16.h>
#include <stddef.h>
#include <stdint.h>

// MI455X / gfx1250, wave32. fp32 WMMA (16x16x4) for all matmuls (batch B=16 ==
// WMMA M; reference is fp32 and the 128-step recurrence punishes bf16 weight
// rounding). Readout GEMM (134 GFLOP, W_out is L2-resident at 131MB < 192MB)
// is one big tiled GEMM using async global->LDS DMA (ASYNCcnt) double-buffered
// against the WMMA pipeline.

typedef float v2f __attribute__((ext_vector_type(2)));
typedef float v8f __attribute__((ext_vector_type(8)));

#define VOCAB 32000
#define DIM   512
#define NE    1024
#define NI    256
#define BATCH 16
#define TLEN  128
#define DTC   0.1f

static __device__ inline v8f wmma4(v2f a, v2f b, v8f c) {
  // (neg_a, A, neg_b, B, c_mod, C, reuse_a, reuse_b)
  return __builtin_amdgcn_wmma_f32_16x16x4_f32(false, a, false, b, (short)0, c,
                                               false, false);
}

// Async DMA: 16 bytes global -> LDS, tracked by ASYNCcnt (no VGPR round-trip).
static __device__ inline void async_b128(unsigned ldsOff, const float* g) {
  asm volatile("global_load_async_to_lds_b128 %0, %1, off"
               :: "v"(ldsOff), "v"(g)
               : "memory");
}
static __device__ inline void wait_async0() {
  asm volatile("s_wait_asynccnt 0x0" ::: "memory");
}

// ---------------------------------------------------------------------------
// Kernel 1: EXT[t*16+b][n] = (emb[tokens[b][t]] + pe[t]) dot W_in[n]
// M=2048, N=1024, K=512. Block = 16x128 tile, 8 waves of 16x16.
// ---------------------------------------------------------------------------
__global__ __launch_bounds__(256) void biga_ext_kernel(
    const int* __restrict__ tokens, const float* __restrict__ emb,
    const float* __restrict__ pe, const float* __restrict__ W_in,
    float* __restrict__ EXT) {
  __shared__ float As[16][33];
  __shared__ float Bs[32][129];
  const int tid = threadIdx.x;
  const int lane = tid & 31;
  const int wv = tid >> 5;
  const int g = lane >> 4, ml = lane & 15;
  const int m0 = ((int)blockIdx.x >> 3) * 16;
  const int n0 = ((int)blockIdx.x & 7) * 128;
  v8f c = {0.f, 0.f, 0.f, 0.f, 0.f, 0.f, 0.f, 0.f};
  for (int k0 = 0; k0 < DIM; k0 += 32) {
    __syncthreads();
#pragma unroll
    for (int j = 0; j < 2; ++j) {
      int idx = tid + 256 * j;
      int i = idx >> 5, k = idx & 31;
      int r = m0 + i;
      int t = r >> 4, b = r & 15;
      int tok = tokens[b * TLEN + t];
      As[i][k] = emb[(size_t)tok * DIM + k0 + k] + pe[t * DIM + k0 + k];
    }
#pragma unroll
    for (int j = 0; j < 16; ++j) {
      int idx = tid + 256 * j;
      int k = idx & 31, n = idx >> 5;
      Bs[k][n] = W_in[(size_t)(n0 + n) * DIM + k0 + k];
    }
    __syncthreads();
#pragma unroll
    for (int kk = 0; kk < 32; kk += 4) {
      v2f a, b;
      a.x = As[ml][kk + 2 * g];
      a.y = As[ml][kk + 2 * g + 1];
      b.x = Bs[kk + 2 * g][wv * 16 + ml];
      b.y = Bs[kk + 2 * g + 1][wv * 16 + ml];
      c = wmma4(a, b, c);
    }
  }
  const int col = n0 + wv * 16 + ml;
#pragma unroll
  for (int r = 0; r < 8; ++r) {
    int m = 8 * g + r;
    EXT[(size_t)(m0 + m) * NE + col] = c[r];
  }
}

// ---------------------------------------------------------------------------
// Kernel 2: one Euler step. 40 blocks:
//   blk 0..31 : E updates (4 groups x 8 column-tiles of 128), up to 4 terms
//   blk 32..39: I updates (4 groups x 2 column-tiles of 128), 2 terms
// All signed inter/intra-group matmul terms fused into one accumulation;
// sign applied while staging weight tiles into LDS.
// ---------------------------------------------------------------------------
__global__ __launch_bounds__(256) void biga_step_kernel(
    const float* ext_t,
    const float* se, const float* si, const float* a1e, const float* a1i,
    const float* a2e, const float* a2i, const float* ge, const float* gi,
    float* se_n, float* si_n, float* a1e_n, float* a1i_n,
    float* a2e_n, float* a2i_n, float* ge_n, float* gi_n,
    const float* S_Wee, const float* S_Wei, const float* S_Wie, const float* S_Wii,
    const float* A1_Wee, const float* A1_Wei, const float* A1_Wie, const float* A1_Wii,
    const float* A2_Wee, const float* A2_Wei, const float* A2_Wie, const float* A2_Wii,
    const float* G_Wee, const float* G_Wei, const float* G_Wie, const float* G_Wii,
    const float* C_S_A1, const float* C_S_A2, const float* C_A1_G, const float* C_A2_G,
    const float* L_A1_A2, const float* L_A2_A1) {
  __shared__ float As[16][33];
  __shared__ float Bs[32][129];
  const int blk = (int)blockIdx.x;
  const float* Ap[4];
  const float* Wp[4];
  int Kt[4];
  float sg[4];
  int nterms, n0, Nout;
  const float* yold;
  float* ynew;
  const float* extp = nullptr;
  if (blk < 32) {
    int grp = blk >> 3;
    n0 = (blk & 7) * 128;
    Nout = NE;
    if (grp == 0) {
      nterms = 2;
      Ap[0] = se;  Wp[0] = S_Wee;  Kt[0] = NE; sg[0] =  1.f;
      Ap[1] = si;  Wp[1] = S_Wei;  Kt[1] = NI; sg[1] = -1.f;
      yold = se; ynew = se_n; extp = ext_t;
    } else if (grp == 1) {
      nterms = 4;
      Ap[0] = a1e; Wp[0] = A1_Wee; Kt[0] = NE; sg[0] =  1.f;
      Ap[1] = a1i; Wp[1] = A1_Wei; Kt[1] = NI; sg[1] = -1.f;
      Ap[2] = se;  Wp[2] = C_S_A1; Kt[2] = NE; sg[2] =  1.f;
      Ap[3] = a2i; Wp[3] = L_A2_A1;Kt[3] = NI; sg[3] = -1.f;
      yold = a1e; ynew = a1e_n;
    } else if (grp == 2) {
      nterms = 4;
      Ap[0] = a2e; Wp[0] = A2_Wee; Kt[0] = NE; sg[0] =  1.f;
      Ap[1] = a2i; Wp[1] = A2_Wei; Kt[1] = NI; sg[1] = -1.f;
      Ap[2] = se;  Wp[2] = C_S_A2; Kt[2] = NE; sg[2] =  1.f;
      Ap[3] = a1i; Wp[3] = L_A1_A2;Kt[3] = NI; sg[3] = -1.f;
      yold = a2e; ynew = a2e_n;
    } else {
      nterms = 4;
      Ap[0] = ge;  Wp[0] = G_Wee;  Kt[0] = NE; sg[0] =  1.f;
      Ap[1] = gi;  Wp[1] = G_Wei;  Kt[1] = NI; sg[1] = -1.f;
      Ap[2] = a1e; Wp[2] = C_A1_G; Kt[2] = NE; sg[2] =  1.f;
      Ap[3] = a2e; Wp[3] = C_A2_G; Kt[3] = NE; sg[3] =  1.f;
      yold = ge; ynew = ge_n;
    }
  } else {
    int grp = (blk - 32) >> 1;
    n0 = ((blk - 32) & 1) * 128;
    Nout = NI;
    nterms = 2;
    Kt[0] = NE; sg[0] = 1.f;
    Kt[1] = NI; sg[1] = -1.f;
    if (grp == 0)      { Ap[0] = se;  Wp[0] = S_Wie;  Ap[1] = si;  Wp[1] = S_Wii;  yold = si;  ynew = si_n; }
    else if (grp == 1) { Ap[0] = a1e; Wp[0] = A1_Wie; Ap[1] = a1i; Wp[1] = A1_Wii; yold = a1i; ynew = a1i_n; }
    else if (grp == 2) { Ap[0] = a2e; Wp[0] = A2_Wie; Ap[1] = a2i; Wp[1] = A2_Wii; yold = a2i; ynew = a2i_n; }
    else               { Ap[0] = ge;  Wp[0] = G_Wie;  Ap[1] = gi;  Wp[1] = G_Wii;  yold = gi;  ynew = gi_n; }
  }
  const int tid = threadIdx.x, lane = tid & 31, wv = tid >> 5;
  const int g = lane >> 4, ml = lane & 15;
  v8f c = {0.f, 0.f, 0.f, 0.f, 0.f, 0.f, 0.f, 0.f};
  for (int ti = 0; ti < nterms; ++ti) {
    const float* A = Ap[ti];
    const float* W = Wp[ti];
    const int K = Kt[ti];
    const float s = sg[ti];
    for (int k0 = 0; k0 < K; k0 += 32) {
      __syncthreads();
#pragma unroll
      for (int j = 0; j < 2; ++j) {
        int idx = tid + 256 * j;
        int i = idx >> 5, k = idx & 31;
        As[i][k] = A[i * K + k0 + k];
      }
#pragma unroll
      for (int j = 0; j < 16; ++j) {
        int idx = tid + 256 * j;
        int k = idx & 31, n = idx >> 5;
        Bs[k][n] = s * W[(size_t)(n0 + n) * K + k0 + k];
      }
      __syncthreads();
#pragma unroll
      for (int kk = 0; kk < 32; kk += 4) {
        v2f a, b;
        a.x = As[ml][kk + 2 * g];
        a.y = As[ml][kk + 2 * g + 1];
        b.x = Bs[kk + 2 * g][wv * 16 + ml];
        b.y = Bs[kk + 2 * g + 1][wv * 16 + ml];
        c = wmma4(a, b, c);
      }
    }
  }
  const int col = n0 + wv * 16 + ml;
#pragma unroll
  for (int r = 0; r < 8; ++r) {
    int m = 8 * g + r;
    float pre = c[r];
    if (extp) pre += extp[m * NE + col];
    float y = yold[m * Nout + col];
    float act = fmaxf(pre, 0.f);
    ynew[m * Nout + col] = y + DTC * (act - y);
  }
}

// ---------------------------------------------------------------------------
// Kernel 3: logits = GE(2048x1024) @ W_out.T(1024x32000), out layout (B,T,V).
// 128x128 block tile, K-chunks of 32, double-buffered async global->LDS DMA
// (b128, ASYNCcnt) overlapping the next tile fetch with the 64 WMMAs of the
// current chunk. Both LDS tiles stored row-major [row][k] with stride 36
// (144B: 16B-aligned for b128, odd dword-quad => conflict-free reads).
// ---------------------------------------------------------------------------
__global__ __launch_bounds__(256) void biga_readout_kernel(
    const float* __restrict__ GEH, const float* __restrict__ W_out,
    float* __restrict__ out) {
  __shared__ float Abuf[2][128][36];
  __shared__ float Bbuf[2][128][36];
  const int tid = threadIdx.x, lane = tid & 31, wv = tid >> 5;
  const int g = lane >> 4, ml = lane & 15;
  const int m0 = ((int)blockIdx.x & 15) * 128;   // 16 M tiles
  const int n0 = ((int)blockIdx.x >> 4) * 128;   // 250 N tiles

  // 128 rows x 32 k-floats per tile = 4096 floats; 4 async b128 per thread.
  auto stage = [&](const float* gbase, int row0, int k0, float (*dst)[36]) {
#pragma unroll
    for (int j = 0; j < 4; ++j) {
      int idx = tid + 256 * j;        // 0..1023
      int i = idx >> 3;               // 0..127  (tile row)
      int kq = (idx & 7) * 4;         // 0..28   (k within chunk, x4 floats)
      const float* gp = gbase + (size_t)(row0 + i) * NE + k0 + kq;
      unsigned l = (unsigned)(uintptr_t)&dst[i][kq];
      async_b128(l, gp);
    }
  };

  const v8f z8 = {0.f, 0.f, 0.f, 0.f, 0.f, 0.f, 0.f, 0.f};
  v8f acc[8];
#pragma unroll
  for (int i = 0; i < 8; ++i) acc[i] = z8;

  stage(GEH,   m0, 0, Abuf[0]);
  stage(W_out, n0, 0, Bbuf[0]);

  const int nchunks = NE / 32;  // 32
  for (int kc = 0; kc < nchunks; ++kc) {
    wait_async0();       // this wave's prefetch for chunk kc has landed in LDS
    __syncthreads();     // ... and every other wave's too; buf reuse is safe
    const int cur = kc & 1;
    if (kc + 1 < nchunks) {
      stage(GEH,   m0, (kc + 1) * 32, Abuf[cur ^ 1]);
      stage(W_out, n0, (kc + 1) * 32, Bbuf[cur ^ 1]);
    }
#pragma unroll
    for (int kk = 0; kk < 32; kk += 4) {
      v2f b;
      b.x = Bbuf[cur][wv * 16 + ml][kk + 2 * g];
      b.y = Bbuf[cur][wv * 16 + ml][kk + 2 * g + 1];
#pragma unroll
      for (int mt = 0; mt < 8; ++mt) {
        v2f a;
        a.x = Abuf[cur][mt * 16 + ml][kk + 2 * g];
        a.y = Abuf[cur][mt * 16 + ml][kk + 2 * g + 1];
        acc[mt] = wmma4(a, b, acc[mt]);
      }
    }
  }

  const int col = n0 + wv * 16 + ml;
#pragma unroll
  for (int mt = 0; mt < 8; ++mt) {
#pragma unroll
    for (int r = 0; r < 8; ++r) {
      int row = m0 + mt * 16 + 8 * g + r;
      int t = row >> 4, b = row & 15;
      out[((size_t)b * TLEN + t) * VOCAB + col] = acc[mt][r];
    }
  }
}

// ---------------------------------------------------------------------------
extern "C" void kernel_launch(void* const* d_in, const int* in_sizes, int n_in,
                              void* d_out, int out_size, void* d_ws, size_t ws_size,
                              hipStream_t stream) {
  (void)in_sizes; (void)n_in; (void)out_size; (void)ws_size;
  const int*   tokens = (const int*)  d_in[0];
  const float* emb    = (const float*)d_in[1];
  const float* pe     = (const float*)d_in[2];
  const float* W_in   = (const float*)d_in[3];
  const float* W_out  = (const float*)d_in[4];
  const float* S_Wee  = (const float*)d_in[5];
  const float* S_Wei  = (const float*)d_in[6];
  const float* S_Wie  = (const float*)d_in[7];
  const float* S_Wii  = (const float*)d_in[8];
  const float* A1_Wee = (const float*)d_in[9];
  const float* A1_Wei = (const float*)d_in[10];
  const float* A1_Wie = (const float*)d_in[11];
  const float* A1_Wii = (const float*)d_in[12];
  const float* A2_Wee = (const float*)d_in[13];
  const float* A2_Wei = (const float*)d_in[14];
  const float* A2_Wie = (const float*)d_in[15];
  const float* A2_Wii = (const float*)d_in[16];
  const float* G_Wee  = (const float*)d_in[17];
  const float* G_Wei  = (const float*)d_in[18];
  const float* G_Wie  = (const float*)d_in[19];
  const float* G_Wii  = (const float*)d_in[20];
  const float* C_S_A1 = (const float*)d_in[21];
  const float* C_S_A2 = (const float*)d_in[22];
  const float* C_A1_G = (const float*)d_in[23];
  const float* C_A2_G = (const float*)d_in[24];
  const float* L_A1_A2= (const float*)d_in[25];
  const float* L_A2_A1= (const float*)d_in[26];

  float* ws = (float*)d_ws;
  size_t o = 0;
  float* EXT = ws + o; o += (size_t)TLEN * BATCH * NE;   // (T*B, NE)
  float* GEH = ws + o; o += (size_t)TLEN * BATCH * NE;   // ge history per step
  float* ZE  = ws + o; o += (size_t)BATCH * NE;          // zero E state
  float* ZI  = ws + o; o += (size_t)BATCH * NI;          // zero I state
  float* SEb[2];  for (int i = 0; i < 2; ++i) { SEb[i]  = ws + o; o += (size_t)BATCH * NE; }
  float* A1Eb[2]; for (int i = 0; i < 2; ++i) { A1Eb[i] = ws + o; o += (size_t)BATCH * NE; }
  float* A2Eb[2]; for (int i = 0; i < 2; ++i) { A2Eb[i] = ws + o; o += (size_t)BATCH * NE; }
  float* SIb[2];  for (int i = 0; i < 2; ++i) { SIb[i]  = ws + o; o += (size_t)BATCH * NI; }
  float* A1Ib[2]; for (int i = 0; i < 2; ++i) { A1Ib[i] = ws + o; o += (size_t)BATCH * NI; }
  float* A2Ib[2]; for (int i = 0; i < 2; ++i) { A2Ib[i] = ws + o; o += (size_t)BATCH * NI; }
  float* GIb[2];  for (int i = 0; i < 2; ++i) { GIb[i]  = ws + o; o += (size_t)BATCH * NI; }

  hipMemsetAsync(ZE, 0, (size_t)BATCH * NE * sizeof(float), stream);
  hipMemsetAsync(ZI, 0, (size_t)BATCH * NI * sizeof(float), stream);

  biga_ext_kernel<<<1024, 256, 0, stream>>>(tokens, emb, pe, W_in, EXT);

  for (int t = 0; t < TLEN; ++t) {
    const int cb = t & 1;          // buffer written at step t
    const int pb = (t + 1) & 1;    // buffer written at step t-1
    const float* se_p  = t ? SEb[pb]  : ZE;
    const float* si_p  = t ? SIb[pb]  : ZI;
    const float* a1e_p = t ? A1Eb[pb] : ZE;
    const float* a1i_p = t ? A1Ib[pb] : ZI;
    const float* a2e_p = t ? A2Eb[pb] : ZE;
    const float* a2i_p = t ? A2Ib[pb] : ZI;
    const float* ge_p  = t ? (GEH + (size_t)(t - 1) * BATCH * NE) : ZE;
    const float* gi_p  = t ? GIb[pb]  : ZI;
    biga_step_kernel<<<40, 256, 0, stream>>>(
        EXT + (size_t)t * BATCH * NE,
        se_p, si_p, a1e_p, a1i_p, a2e_p, a2i_p, ge_p, gi_p,
        SEb[cb], SIb[cb], A1Eb[cb], A1Ib[cb], A2Eb[cb], A2Ib[cb],
        GEH + (size_t)t * BATCH * NE, GIb[cb],
        S_Wee, S_Wei, S_Wie, S_Wii,
        A1_Wee, A1_Wei, A1_Wie, A1_Wii,
        A2_Wee, A2_Wei, A2_Wie, A2_Wii,
        G_Wee, G_Wei, G_Wie, G_Wii,
        C_S_A1, C_S_A2, C_A1_G, C_A2_G, L_A1_A2, L_A2_A1);
  }

  biga_readout_kernel<<<16 * 250, 256, 0, stream>>>(GEH, W_out, (float*)d_out);
}